// Channel_attention_19172734009843
// MI455X (gfx1250) — compile-verified
//
#include <hip/hip_runtime.h>

#define BATCH 64
#define CH    64
#define TT    4000
#define TK    80             // T-chunk staged in LDS
#define NCHUNK (TT / TK)     // 50
#define XS_S  84             // LDS row stride (floats): 336B = 21*16 -> 16B-aligned rows
#define XBUF  (CH * XS_S)    // floats per buffer (5376)
#define GS_S  72             // LDS row stride for G / attention (floats)
#define EPSV  1e-8f

typedef __attribute__((ext_vector_type(2))) float v2f;
typedef __attribute__((ext_vector_type(8))) float v8f;
typedef __attribute__((ext_vector_type(4))) int   v4i;

// ---------------- CDNA5 async global->LDS copy path ----------------
#if __has_builtin(__builtin_amdgcn_global_load_async_to_lds_b128)
#define ASYNC_OK 1
// builtin expects: (int4 addrspace(1)* src, int4 addrspace(3)* dst, imm offset, imm cpol)
typedef __attribute__((address_space(1))) v4i* gv4p;
typedef __attribute__((address_space(3))) v4i* lv4p;
__device__ __forceinline__ void async_cp16(const float* g, float* l) {
    // 16B global -> LDS, tracked by ASYNCcnt (GLOBAL_LOAD_ASYNC_TO_LDS_B128)
    __builtin_amdgcn_global_load_async_to_lds_b128((gv4p)g, (lv4p)l, 0, 0);
}
__device__ __forceinline__ void wait_async() {
#if __has_builtin(__builtin_amdgcn_s_wait_asynccnt)
    __builtin_amdgcn_s_wait_asynccnt(0);
#else
    asm volatile("s_wait_asynccnt 0" ::: "memory");
#endif
}
#else
#define ASYNC_OK 0
__device__ __forceinline__ void wait_async() {}
#endif

__device__ __forceinline__ v8f wmma_f32(v2f a, v2f b, v8f c) {
    // D = A(16x4) * B(4x16) + C, f32 everywhere.
    return __builtin_amdgcn_wmma_f32_16x16x4_f32(
        /*neg_a=*/false, a, /*neg_b=*/false, b,
        /*c_mod=*/(short)0, c, /*reuse_a=*/false, /*reuse_b=*/false);
}

// cooperative chunk loader: 64 rows x 80 cols (1280 float4, 5 per thread)
__device__ __forceinline__ void load_chunk(const float* __restrict__ xb,
                                           float* __restrict__ dst,
                                           int t0, int tid) {
#pragma unroll
    for (int it = 0; it < 5; ++it) {
        const int fi  = tid + it * 256;     // 0..1279
        const int row = fi / 20;            // 20 float4 per row
        const int c4  = fi - row * 20;
#if ASYNC_OK
        async_cp16(xb + row * TT + t0 + c4 * 4, dst + row * XS_S + c4 * 4);
#else
        const float4 v = *reinterpret_cast<const float4*>(xb + row * TT + t0 + c4 * 4);
        *reinterpret_cast<float4*>(dst + row * XS_S + c4 * 4) = v;
#endif
    }
}

__global__ __launch_bounds__(256)
void chan_attn_kernel(const float* __restrict__ x,
                      const float* __restrict__ w1, const float* __restrict__ b1,
                      const float* __restrict__ w2, const float* __restrict__ b2,
                      const float* __restrict__ gammap,
                      float* __restrict__ out)
{
    __shared__ float Xs[2 * XBUF];    // 43,008 B : double-buffered X chunk
    __shared__ float Gs[CH * GS_S];   // 18,432 B : Gram -> energy -> attention
    __shared__ float Srow[CH];        // row sums

    const int tid  = threadIdx.x;
    const int lane = tid & 31;
    const int wave = tid >> 5;
    const int half = lane >> 4;   // 0: lanes 0-15, 1: lanes 16-31
    const int l16  = lane & 15;
    const int b    = blockIdx.x;

    const float* __restrict__ xb = x   + (size_t)b * CH * TT;
    float*       __restrict__ ob = out + (size_t)b * CH * TT;

    // ---- fold the two 1x1 convs into 4 scalars ----
    float alpha = 0.f, beta = 0.f, g2 = 0.f, dcon = 0.f;
#pragma unroll
    for (int j = 0; j < 8; ++j) {
        const float a1 = w1[j], a2 = w2[j], c1 = b1[j], c2 = b2[j];
        alpha += a1 * a2; beta += a1 * c2; g2 += c1 * a2; dcon += c1 * c2;
    }
    dcon *= (float)TT;
    const float gamma = gammap[0];

    // ---- Phase 1: Gram matrix G = X * X^T, K = 4000, double-buffered ----
    const int tA = wave * 2;
    const int iA = tA >> 2,       jA = tA & 3;
    const int iB = (tA + 1) >> 2, jB = (tA + 1) & 3;
    v8f accA = {0.f,0.f,0.f,0.f,0.f,0.f,0.f,0.f};
    v8f accB = {0.f,0.f,0.f,0.f,0.f,0.f,0.f,0.f};
    float rowsum = 0.f;

    load_chunk(xb, Xs, 0, tid);
    wait_async();
    __syncthreads();

    for (int ch = 0; ch < NCHUNK; ++ch) {
        float* cur = Xs + (ch & 1) * XBUF;
        float* nxt = Xs + ((ch & 1) ^ 1) * XBUF;
        if (ch + 1 < NCHUNK)                         // prefetch next chunk (async)
            load_chunk(xb, nxt, (ch + 1) * TK, tid);

        // row-sum partials (waves 0-1, wave-uniform predicate)
        if (tid < CH) {
            float s = 0.f;
            for (int c4 = 0; c4 < TK / 4; ++c4) {
                const float4 v = *reinterpret_cast<const float4*>(&cur[tid * XS_S + c4 * 4]);
                s += v.x + v.y + v.z + v.w;
            }
            rowsum += s;
        }

        // Gram accumulation: A[m][k]=X[i16+m][kk+k], B[k][n]=X[j16+n][kk+k]
        for (int kk = 0; kk < TK; kk += 4) {
            const int colA = kk + 2 * half;
            const v2f a0  = *reinterpret_cast<const v2f*>(&cur[(iA * 16 + l16) * XS_S + colA]);
            const v2f b0  = *reinterpret_cast<const v2f*>(&cur[(jA * 16 + l16) * XS_S + colA]);
            accA = wmma_f32(a0, b0, accA);
            const v2f a1  = *reinterpret_cast<const v2f*>(&cur[(iB * 16 + l16) * XS_S + colA]);
            const v2f b1v = *reinterpret_cast<const v2f*>(&cur[(jB * 16 + l16) * XS_S + colA]);
            accB = wmma_f32(a1, b1v, accB);
        }

        wait_async();          // next chunk fully landed in LDS
        __syncthreads();       // all waves done reading `cur`
    }

    if (tid < CH) Srow[tid] = rowsum;
    // write Gram tiles: D vgpr r -> rows r (lanes 0-15) / r+8 (lanes 16-31)
#pragma unroll
    for (int r = 0; r < 8; ++r) {
        Gs[(iA * 16 + r + 8 * half) * GS_S + jA * 16 + l16] = accA[r];
        Gs[(iB * 16 + r + 8 * half) * GS_S + jB * 16 + l16] = accB[r];
    }
    __syncthreads();

    // ---- Phase 2: energy fold + min-max norm + softmax (one row per thread<64) ----
    if (tid < CH) {
        const float sc = Srow[tid];
        float mx = -3.4e38f, mn = 3.4e38f;
        for (int e = 0; e < CH; ++e) {
            const float en = alpha * Gs[tid * GS_S + e] + beta * sc + g2 * Srow[e] + dcon;
            Gs[tid * GS_S + e] = en;
            mx = fmaxf(mx, en); mn = fminf(mn, en);
        }
        const float inv = 1.f / (mx - mn + EPSV);
        float sum = 0.f;
        for (int e = 0; e < CH; ++e) {
            const float p  = (Gs[tid * GS_S + e] - mn) * inv;   // in [0,1]
            const float ex = expf(p - 1.0f);                    // max-shifted softmax
            Gs[tid * GS_S + e] = ex;
            sum += ex;
        }
        const float isum = 1.f / sum;
        for (int e = 0; e < CH; ++e) Gs[tid * GS_S + e] *= isum;
    }
    __syncthreads();

    // ---- Phase 3: out[c][t] = gamma * sum_e att[e][c] * X[e][t] + x[c][t] ----
    // M=64 (c), N=4000 (t), K=64 (e); 20 tiles/chunk, round-robin over waves.
    load_chunk(xb, Xs, 0, tid);
    wait_async();
    __syncthreads();

    for (int ch = 0; ch < NCHUNK; ++ch) {
        const int t0 = ch * TK;
        float* cur = Xs + (ch & 1) * XBUF;
        float* nxt = Xs + ((ch & 1) ^ 1) * XBUF;
        if (ch + 1 < NCHUNK)
            load_chunk(xb, nxt, (ch + 1) * TK, tid);

        for (int tt = wave; tt < 20; tt += 8) {     // wave-uniform trip count
            const int i   = tt / 5;                 // row tile (c block)
            const int jt  = tt - i * 5;             // col tile within chunk
            const int c0  = i * 16;
            const int tc0 = jt * 16;
            v8f acc = {0.f,0.f,0.f,0.f,0.f,0.f,0.f,0.f};
            for (int ks = 0; ks < 16; ++ks) {
                const int er = ks * 4 + 2 * half;
                v2f af, bf;
                // A[m][k] = att[e0+k][c0+m]  (transposed attention)
                af.x = Gs[er * GS_S + c0 + l16];
                af.y = Gs[(er + 1) * GS_S + c0 + l16];
                // B[k][n] = X[e0+k][tc0+n]
                bf.x = cur[er * XS_S + tc0 + l16];
                bf.y = cur[(er + 1) * XS_S + tc0 + l16];
                acc = wmma_f32(af, bf, acc);
            }
#pragma unroll
            for (int r = 0; r < 8; ++r) {
                const int row = c0 + r + 8 * half;
                const int col = tc0 + l16;
                ob[row * TT + t0 + col] = gamma * acc[r] + cur[row * XS_S + col];
            }
        }

        wait_async();
        __syncthreads();
    }
}

extern "C" void kernel_launch(void* const* d_in, const int* in_sizes, int n_in,
                              void* d_out, int out_size, void* d_ws, size_t ws_size,
                              hipStream_t stream) {
    const float* x  = (const float*)d_in[0];
    const float* w1 = (const float*)d_in[1];
    const float* b1 = (const float*)d_in[2];
    const float* w2 = (const float*)d_in[3];
    const float* b2 = (const float*)d_in[4];
    const float* gm = (const float*)d_in[5];
    float* out = (float*)d_out;
    hipLaunchKernelGGL(chan_attn_kernel, dim3(BATCH), dim3(256), 0, stream,
                       x, w1, b1, w2, b2, gm, out);
}